// GATSuper_6030134083941
// MI455X (gfx1250) — compile-verified
//
#include <hip/hip_runtime.h>
#include <math.h>

// Problem sizes (fixed by the reference)
#define NN   50000
#define EE   800000
#define FINX 128        // encoder input features
#define HIDC 64
#define NH   4
#define OUTC 40
#define NG   8
#define DD   256        // HID*H
#define EP   (EE + NN)  // edges incl. self-loops = 850000
#define TILES (NN / 16) // 3125 exactly
#define NEG  0.2f

typedef __attribute__((ext_vector_type(2))) float v2f;
typedef __attribute__((ext_vector_type(8))) float v8f;

// D = A(16x4,f32) * B(4x16,f32) + C(16x16,f32)  -- CDNA5 fp32 WMMA
__device__ __forceinline__ v8f wmma4(v2f a, v2f b, v8f c) {
  return __builtin_amdgcn_wmma_f32_16x16x4_f32(false, a, false, b, (short)0, c,
                                               false, false);
}

// ASYNCcnt-tracked global->LDS copy (CDNA5). lds_byte = LDS byte offset
// (low 32 bits of the shared-aperture flat address).
__device__ __forceinline__ void async_g2l_b128(unsigned lds_byte,
                                               const void* gaddr) {
  asm volatile("global_load_async_to_lds_b128 %0, %1, off"
               :: "v"(lds_byte), "v"(gaddr) : "memory");
}
__device__ __forceinline__ void wait_async0() {
  asm volatile("s_wait_asynccnt 0x0" ::: "memory");
}
__device__ __forceinline__ unsigned lds_off(const void* p) {
  return (unsigned)(unsigned long long)p;  // low 32 bits = LDS offset
}

// float atomic-max via sign trick (dest initialized to -inf)
__device__ __forceinline__ void atomicMaxF(float* addr, float v) {
  if (v >= 0.f) atomicMax((int*)addr, __float_as_int(v));
  else          atomicMin((unsigned int*)addr, __float_as_uint(v));
}

__device__ __forceinline__ void edge_sd(const int* __restrict__ ei, int e,
                                        int& s, int& d) {
  if (e < EE) { s = ei[e]; d = ei[EE + e]; }
  else        { s = e - EE; d = s; }       // appended self-loops
}

// ---------------------------------------------------------------------------
// Encoder: h = relu(LN(x@W1 + b1)) @ W2 + b2    (one 16-row tile per wave)
// Padded LDS row strides (=4 mod 64) -> conflict-free WMMA fragment reads.
// ---------------------------------------------------------------------------
#define XST (FINX + 4)   // 132
#define LST (HIDC + 4)   // 68
__global__ void __launch_bounds__(128) enc_kernel(
    const float* __restrict__ x,
    const float* __restrict__ w1, const float* __restrict__ b1,
    const float* __restrict__ gn, const float* __restrict__ be,
    const float* __restrict__ w2, const float* __restrict__ b2,
    float* __restrict__ hout) {
  __shared__ __attribute__((aligned(16))) float xT[4][16 * XST];
  __shared__ __attribute__((aligned(16))) float lnT[4][16 * LST];
  const int wid = threadIdx.x >> 5, lane = threadIdx.x & 31;
  const int lo = lane & 15, hi = lane >> 4;
  const int tile = blockIdx.x * 4 + wid;
  if (tile >= TILES) return;                 // no block barriers used
  const int row0 = tile * 16;

  { // async-stage 16x128 A tile into padded rows
    for (int idx = lane; idx < 16 * (FINX / 4); idx += 32) {
      const int r = idx >> 5, c = idx & 31;  // FINX/4 = 32 float4 per row
      async_g2l_b128(lds_off(&xT[wid][r * XST + c * 4]),
                     x + (size_t)(row0 + r) * FINX + c * 4);
    }
    wait_async0();
  }

  v8f acc[4] = {};
  for (int kk = 0; kk < FINX / 4; ++kk) {
    v2f a = *(const v2f*)&xT[wid][lo * XST + kk * 4 + 2 * hi];
    const int kb = kk * 4 + 2 * hi;
#pragma unroll
    for (int nt = 0; nt < 4; ++nt) {
      v2f b;
      b.x = w1[kb * HIDC + nt * 16 + lo];
      b.y = w1[(kb + 1) * HIDC + nt * 16 + lo];
      acc[nt] = wmma4(a, b, acc[nt]);
    }
  }
  // +bias into LDS for row-wise LayerNorm
#pragma unroll
  for (int nt = 0; nt < 4; ++nt)
#pragma unroll
    for (int j = 0; j < 8; ++j) {
      const int r = j + 8 * hi, c = nt * 16 + lo;
      lnT[wid][r * LST + c] = acc[nt][j] + b1[c];
    }
  // LN + ReLU: lanes 0..15 each own one row (same-wave DS ops are in-order)
  if (lane < 16) {
    float m = 0.f;
    for (int c = 0; c < HIDC; ++c) m += lnT[wid][lane * LST + c];
    m *= (1.f / HIDC);
    float v = 0.f;
    for (int c = 0; c < HIDC; ++c) {
      float d = lnT[wid][lane * LST + c] - m; v += d * d;
    }
    v *= (1.f / HIDC);
    const float rs = rsqrtf(v + 1e-5f);
    for (int c = 0; c < HIDC; ++c) {
      float y = gn[c] * (lnT[wid][lane * LST + c] - m) * rs + be[c];
      lnT[wid][lane * LST + c] = y > 0.f ? y : 0.f;
    }
  }
  // second GEMM: (16x64) @ (64x64)
  v8f acc2[4] = {};
  for (int kk = 0; kk < HIDC / 4; ++kk) {
    v2f a = *(const v2f*)&lnT[wid][lo * LST + kk * 4 + 2 * hi];
    const int kb = kk * 4 + 2 * hi;
#pragma unroll
    for (int nt = 0; nt < 4; ++nt) {
      v2f b;
      b.x = w2[kb * HIDC + nt * 16 + lo];
      b.y = w2[(kb + 1) * HIDC + nt * 16 + lo];
      acc2[nt] = wmma4(a, b, acc2[nt]);
    }
  }
#pragma unroll
  for (int nt = 0; nt < 4; ++nt)
#pragma unroll
    for (int j = 0; j < 8; ++j) {
      const int r = j + 8 * hi, c = nt * 16 + lo;
      hout[(size_t)(row0 + r) * HIDC + c] = acc2[nt][j] + b2[c];
    }
}

// ---------------------------------------------------------------------------
// GAT node GEMM: feat = h_in @ W  ([N,FIN]@[FIN,256]).
// 4 waves/block, wave per 16-row tile (async-staged A, padded stride FIN+4),
// W staged cooperatively in LDS 16-k-row chunks (padded stride DD+16) so the
// 256KB weight matrix is pulled from L2 once per *block*, not once per wave.
// ---------------------------------------------------------------------------
template <int FIN>
__global__ void __launch_bounds__(128) gat_gemm_kernel(
    const float* __restrict__ hin, const float* __restrict__ W,
    float* __restrict__ feat) {
  constexpr int AST = FIN + 4;   // ==4 mod 64 -> conflict-free A frags
  constexpr int WST = DD + 16;   // ==16 mod 64 -> conflict-free B frags
  __shared__ __attribute__((aligned(16))) float aT[4][16 * AST];
  __shared__ __attribute__((aligned(16))) float wT[16 * WST];
  const int wid = threadIdx.x >> 5, lane = threadIdx.x & 31;
  const int lo = lane & 15, hi = lane >> 4;
  const int tile = blockIdx.x * 4 + wid;
  const int tileC = tile < TILES ? tile : TILES - 1;  // keep barriers uniform
  const int row0 = tileC * 16;

  { // async-stage this wave's 16xFIN A tile
    constexpr int V4R = FIN / 4;
    for (int idx = lane; idx < 16 * V4R; idx += 32) {
      const int r = idx / V4R, c = idx % V4R;
      async_g2l_b128(lds_off(&aT[wid][r * AST + c * 4]),
                     hin + (size_t)(row0 + r) * FIN + c * 4);
    }
  }

  v8f acc[16] = {};
  for (int kc = 0; kc < FIN / 16; ++kc) {
    __syncthreads();  // all waves done reading previous W chunk
    for (int idx = threadIdx.x; idx < 16 * 64; idx += 128) {
      const int r = idx >> 6, c = idx & 63;
      *(float4*)(wT + r * WST + c * 4) =
          *(const float4*)(W + (size_t)(kc * 16 + r) * DD + c * 4);
    }
    if (kc == 0) wait_async0();  // A tile resident before first use
    __syncthreads();
#pragma unroll
    for (int k4 = 0; k4 < 4; ++k4) {
      v2f a = *(const v2f*)&aT[wid][lo * AST + kc * 16 + k4 * 4 + 2 * hi];
      const int kb = k4 * 4 + 2 * hi;
#pragma unroll
      for (int nt = 0; nt < 16; ++nt) {
        v2f b;
        b.x = wT[kb * WST + nt * 16 + lo];
        b.y = wT[(kb + 1) * WST + nt * 16 + lo];
        acc[nt] = wmma4(a, b, acc[nt]);
      }
    }
  }
  if (tile < TILES) {
#pragma unroll
    for (int nt = 0; nt < 16; ++nt)
#pragma unroll
      for (int j = 0; j < 8; ++j)
        feat[(size_t)(row0 + j + 8 * hi) * DD + nt * 16 + lo] = acc[nt][j];
  }
}

// attention logits per (node, head): al = <feat[n,h,:], a[h,:]>
__global__ void al_kernel(const float* __restrict__ feat,
                          const float* __restrict__ asrc,
                          const float* __restrict__ adst,
                          float* __restrict__ als, float* __restrict__ ald) {
  const int t = blockIdx.x * blockDim.x + threadIdx.x;
  if (t >= NN * NH) return;
  const int n = t >> 2, h = t & 3;
  const float* f = feat + (size_t)n * DD + h * HIDC;
  const float* as = asrc + h * HIDC;
  const float* ad = adst + h * HIDC;
  float s1 = 0.f, s2 = 0.f;
  for (int c = 0; c < HIDC; ++c) { float v = f[c]; s1 += v * as[c]; s2 += v * ad[c]; }
  als[t] = s1; ald[t] = s2;
}

__global__ void init_kernel(float* __restrict__ acc, float* __restrict__ m,
                            float* __restrict__ s) {
  const size_t i = (size_t)blockIdx.x * blockDim.x + threadIdx.x;
  if (i < (size_t)NN * DD) acc[i] = 0.f;
  if (i < (size_t)NN * NH) { m[i] = -__builtin_inff(); s[i] = 0.f; }
}

__global__ void edge_max_kernel(const int* __restrict__ ei,
                                const float* __restrict__ als,
                                const float* __restrict__ ald,
                                float* __restrict__ m) {
  const int e = blockIdx.x * blockDim.x + threadIdx.x;
  if (e >= EP) return;
  int s, d; edge_sd(ei, e, s, d);
#pragma unroll
  for (int h = 0; h < NH; ++h) {
    float v = als[s * NH + h] + ald[d * NH + h];
    v = v > 0.f ? v : NEG * v;
    atomicMaxF(&m[d * NH + h], v);
  }
}

__global__ void edge_expsum_kernel(const int* __restrict__ ei,
                                   const float* __restrict__ als,
                                   const float* __restrict__ ald,
                                   const float* __restrict__ m,
                                   float* __restrict__ ssum,
                                   float* __restrict__ eb) {
  const int e = blockIdx.x * blockDim.x + threadIdx.x;
  if (e >= EP) return;
  int s, d; edge_sd(ei, e, s, d);
#pragma unroll
  for (int h = 0; h < NH; ++h) {
    float v = als[s * NH + h] + ald[d * NH + h];
    v = v > 0.f ? v : NEG * v;
    const float ex = __expf(v - m[d * NH + h]);
    eb[(size_t)e * NH + h] = ex;
    atomicAdd(&ssum[d * NH + h], ex);
  }
}

// wave per edge: out[dst,:] += e_uv * feat[src,:]  (normalization deferred)
__global__ void __launch_bounds__(256) edge_agg_kernel(
    const int* __restrict__ ei, const float* __restrict__ feat,
    const float* __restrict__ eb, float* __restrict__ acc) {
  const int wid = threadIdx.x >> 5, lane = threadIdx.x & 31;
  const int e = blockIdx.x * 8 + wid;
  if (e >= EP) return;
  int s, d; edge_sd(ei, e, s, d);
  const float alpha = eb[(size_t)e * NH + (lane >> 3)];
  const float4* fs = (const float4*)(feat + (size_t)s * DD + lane * 8);
  float f[8];
  *(float4*)&f[0] = fs[0];
  *(float4*)&f[4] = fs[1];
  float* out = acc + (size_t)d * DD + lane * 8;
#pragma unroll
  for (int k = 0; k < 8; ++k) atomicAdd(out + k, f[k] * alpha);
}

// h_out = elu(acc / (s[dst] + 1e-16) + bias)   (in place)
__global__ void finalize_kernel(float* __restrict__ acc,
                                const float* __restrict__ s,
                                const float* __restrict__ bias) {
  const size_t i = (size_t)blockIdx.x * blockDim.x + threadIdx.x;
  if (i >= (size_t)NN * DD) return;
  const int n = (int)(i >> 8), c = (int)(i & 255), h = c >> 6;
  const float v = acc[i] / (s[n * NH + h] + 1e-16f) + bias[c];
  acc[i] = v > 0.f ? v : (__expf(v) - 1.f);
}

__global__ void pool_init(float* __restrict__ psum, float* __restrict__ pcnt) {
  for (int i = threadIdx.x; i < NG * DD; i += 256) psum[i] = 0.f;
  if (threadIdx.x < NG) pcnt[threadIdx.x] = 0.f;
}

// hierarchical mean-pool: thread t exclusively owns feature column t
__global__ void __launch_bounds__(256) pool_kernel(const float* __restrict__ h,
                                                   const int* __restrict__ batch,
                                                   float* __restrict__ psum,
                                                   float* __restrict__ pcnt) {
  __shared__ float ls[NG * DD];
  __shared__ int lc[NG];
  const int t = threadIdx.x;
  for (int i = t; i < NG * DD; i += 256) ls[i] = 0.f;
  if (t < NG) lc[t] = 0;
  __syncthreads();
  const int n0 = blockIdx.x * 1024;
  const int n1 = (n0 + 1024 < NN) ? n0 + 1024 : NN;
  for (int n = n0; n < n1; ++n) {
    const int g = batch[n];
    ls[g * DD + t] += h[(size_t)n * DD + t];
    if (t == 0) lc[g]++;
  }
  __syncthreads();
  for (int i = t; i < NG * DD; i += 256) atomicAdd(&psum[i], ls[i]);
  if (t < NG) atomicAdd(&pcnt[t], (float)lc[t]);
}

// tiny decoder on [8,256]: single block
__global__ void __launch_bounds__(256) decoder_kernel(
    const float* __restrict__ psum, const float* __restrict__ pcnt,
    const float* __restrict__ w1, const float* __restrict__ b1,
    const float* __restrict__ gn, const float* __restrict__ be,
    const float* __restrict__ w2, const float* __restrict__ b2,
    float* __restrict__ out) {
  __shared__ float pooled[NG * DD];
  __shared__ float z[NG * HIDC];
  const int t = threadIdx.x;
  for (int i = t; i < NG * DD; i += 256) {
    const int g = i >> 8;
    float c = pcnt[g]; c = c > 1.f ? c : 1.f;
    pooled[i] = psum[i] / c;
  }
  __syncthreads();
  for (int idx = t; idx < NG * HIDC; idx += 256) {
    const int r = idx >> 6, c = idx & 63;
    float a = b1[c];
    for (int k = 0; k < DD; ++k) a += pooled[r * DD + k] * w1[k * HIDC + c];
    z[idx] = a;
  }
  __syncthreads();
  if (t < NG) {
    float m = 0.f;
    for (int c = 0; c < HIDC; ++c) m += z[t * HIDC + c];
    m /= HIDC;
    float v = 0.f;
    for (int c = 0; c < HIDC; ++c) { float d = z[t * HIDC + c] - m; v += d * d; }
    v /= HIDC;
    const float rs = rsqrtf(v + 1e-5f);
    for (int c = 0; c < HIDC; ++c) {
      float y = gn[c] * (z[t * HIDC + c] - m) * rs + be[c];
      z[t * HIDC + c] = y > 0.f ? y : 0.f;
    }
  }
  __syncthreads();
  for (int idx = t; idx < NG * OUTC; idx += 256) {
    const int r = idx / OUTC, c = idx % OUTC;
    float a = b2[c];
    for (int k = 0; k < HIDC; ++k) a += z[r * HIDC + k] * w2[k * OUTC + c];
    out[idx] = a;
  }
}

// ---------------------------------------------------------------------------
static void gat_tail(const float* feat, const float* as, const float* ad,
                     const float* bias, const int* ei, float* als, float* ald,
                     float* mb, float* sb, float* eb, float* accOut,
                     hipStream_t stream) {
  al_kernel<<<(NN * NH + 255) / 256, 256, 0, stream>>>(feat, as, ad, als, ald);
  init_kernel<<<(int)(((size_t)NN * DD + 255) / 256), 256, 0, stream>>>(accOut, mb, sb);
  edge_max_kernel<<<(EP + 255) / 256, 256, 0, stream>>>(ei, als, ald, mb);
  edge_expsum_kernel<<<(EP + 255) / 256, 256, 0, stream>>>(ei, als, ald, mb, sb, eb);
  edge_agg_kernel<<<(EP + 7) / 8, 256, 0, stream>>>(ei, feat, eb, accOut);
  finalize_kernel<<<(int)(((size_t)NN * DD + 255) / 256), 256, 0, stream>>>(accOut, sb, bias);
}

extern "C" void kernel_launch(void* const* d_in, const int* in_sizes, int n_in,
                              void* d_out, int out_size, void* d_ws,
                              size_t ws_size, hipStream_t stream) {
  const float* x       = (const float*)d_in[0];
  const int*   ei      = (const int*)d_in[1];
  const int*   batch   = (const int*)d_in[2];
  const float* enc_w1  = (const float*)d_in[3];
  const float* enc_b1  = (const float*)d_in[4];
  const float* enc_g   = (const float*)d_in[5];
  const float* enc_be  = (const float*)d_in[6];
  const float* enc_w2  = (const float*)d_in[7];
  const float* enc_b2  = (const float*)d_in[8];
  const float* conv_w[3]  = {(const float*)d_in[9],  (const float*)d_in[13], (const float*)d_in[17]};
  const float* conv_as[3] = {(const float*)d_in[10], (const float*)d_in[14], (const float*)d_in[18]};
  const float* conv_ad[3] = {(const float*)d_in[11], (const float*)d_in[15], (const float*)d_in[19]};
  const float* conv_b[3]  = {(const float*)d_in[12], (const float*)d_in[16], (const float*)d_in[20]};
  const float* dec_w1  = (const float*)d_in[21];
  const float* dec_b1  = (const float*)d_in[22];
  const float* dec_g   = (const float*)d_in[23];
  const float* dec_be  = (const float*)d_in[24];
  const float* dec_w2  = (const float*)d_in[25];
  const float* dec_b2  = (const float*)d_in[26];

  // workspace layout (~183 MB of f32)
  float* p    = (float*)d_ws;
  float* hEnc = p; p += (size_t)NN * HIDC;
  float* bufA = p; p += (size_t)NN * DD;
  float* bufB = p; p += (size_t)NN * DD;
  float* feat = p; p += (size_t)NN * DD;
  float* als  = p; p += (size_t)NN * NH;
  float* ald  = p; p += (size_t)NN * NH;
  float* mb   = p; p += (size_t)NN * NH;
  float* sb   = p; p += (size_t)NN * NH;
  float* eb   = p; p += (size_t)EP * NH;
  float* psum = p; p += NG * DD;
  float* pcnt = p; p += NG;

  enc_kernel<<<(TILES + 3) / 4, 128, 0, stream>>>(x, enc_w1, enc_b1, enc_g,
                                                  enc_be, enc_w2, enc_b2, hEnc);
  // layer 0: [N,64] -> bufA
  gat_gemm_kernel<HIDC><<<(TILES + 3) / 4, 128, 0, stream>>>(hEnc, conv_w[0], feat);
  gat_tail(feat, conv_as[0], conv_ad[0], conv_b[0], ei, als, ald, mb, sb, eb, bufA, stream);
  // layer 1: bufA -> bufB
  gat_gemm_kernel<DD><<<(TILES + 3) / 4, 128, 0, stream>>>(bufA, conv_w[1], feat);
  gat_tail(feat, conv_as[1], conv_ad[1], conv_b[1], ei, als, ald, mb, sb, eb, bufB, stream);
  // layer 2: bufB -> bufA
  gat_gemm_kernel<DD><<<(TILES + 3) / 4, 128, 0, stream>>>(bufB, conv_w[2], feat);
  gat_tail(feat, conv_as[2], conv_ad[2], conv_b[2], ei, als, ald, mb, sb, eb, bufA, stream);

  pool_init<<<1, 256, 0, stream>>>(psum, pcnt);
  pool_kernel<<<(NN + 1023) / 1024, 256, 0, stream>>>(bufA, batch, psum, pcnt);
  decoder_kernel<<<1, 256, 0, stream>>>(psum, pcnt, dec_w1, dec_b1, dec_g,
                                        dec_be, dec_w2, dec_b2, (float*)d_out);
  (void)in_sizes; (void)n_in; (void)out_size; (void)ws_size;
}